// UFGAT_14577119003214
// MI455X (gfx1250) — compile-verified
//
#include <hip/hip_runtime.h>
#include <hip/hip_bf16.h>
#include <cstddef>

typedef __attribute__((ext_vector_type(2))) float v2f;
typedef __attribute__((ext_vector_type(8))) float v8f;

#define HID 64
#define NEG_SLOPE 0.2f

__device__ __forceinline__ float leaky(float x) { return x > 0.f ? x : NEG_SLOPE * x; }

// monotone float->uint key so unsigned atomicMax == float max
__device__ __forceinline__ unsigned fkey(float f) {
    unsigned u = __float_as_uint(f);
    return (u & 0x80000000u) ? ~u : (u | 0x80000000u);
}
__device__ __forceinline__ float kinv(unsigned k) {
    unsigned u = (k & 0x80000000u) ? (k & 0x7fffffffu) : ~k;
    return __uint_as_float(u);
}

#define WMMA_F32X4(a, b, c) \
    __builtin_amdgcn_wmma_f32_16x16x4_f32(false, (a), false, (b), (short)0, (c), false, false)

// ---------------------------------------------------------------------------
// GEMM: H[M,64] = A[M,K] @ W[K,64], fp32 WMMA 16x16x4.
// block = 128 threads (4 waves). Block handles 64 rows x 64 cols:
// wave w -> cols 16w..16w+15, 4 row-tiles of 16 (B fragments reused 4x).
// ---------------------------------------------------------------------------
__global__ __launch_bounds__(128) void gemm_n64_wmma(const float* __restrict__ A,
                                                     const float* __restrict__ W,
                                                     float* __restrict__ H, int M, int K) {
    const int lane  = threadIdx.x & 31;
    const int wave  = threadIdx.x >> 5;
    const int row0  = blockIdx.x * 64;
    const int col0  = wave * 16;
    const int mrow  = lane & 15;
    const int khalf = (lane >> 4) << 1;  // 0 or 2

    const float* arow[4];
#pragma unroll
    for (int t = 0; t < 4; ++t) {
        int ri = row0 + t * 16 + mrow;
        if (ri >= M) ri = M - 1;  // clamp; duplicate compute, stores guarded below
        arow[t] = A + (size_t)ri * K + khalf;
    }
    const float* wcol = W + col0 + mrow;

    v8f acc[4] = {};
    for (int k0 = 0; k0 < K; k0 += 4) {
        v2f b;
        b.x = wcol[(size_t)(k0 + khalf) * HID];
        b.y = wcol[(size_t)(k0 + khalf + 1) * HID];
#pragma unroll
        for (int t = 0; t < 4; ++t) {
            v2f a = *(const v2f*)(arow[t] + k0);
            acc[t] = WMMA_F32X4(a, b, acc[t]);
        }
    }

    const int rbase = (lane >> 4) * 8;
#pragma unroll
    for (int t = 0; t < 4; ++t) {
#pragma unroll
        for (int r = 0; r < 8; ++r) {
            int row = row0 + t * 16 + rbase + r;
            if (row < M) H[(size_t)row * HID + col0 + mrow] = acc[t][r];
        }
    }
}

// ---------------------------------------------------------------------------
// per-node: s_src = h.a_s, s_dst = h.a_d, self-loop score, init segment-max key
// ---------------------------------------------------------------------------
__global__ void node_dots(const float* __restrict__ h, const float* __restrict__ a_s,
                          const float* __restrict__ a_d, float* __restrict__ s_src,
                          float* __restrict__ s_dst, float* __restrict__ selfv,
                          unsigned* __restrict__ mkey, int N) {
    int n = blockIdx.x * blockDim.x + threadIdx.x;
    if (n >= N) return;
    const float* r = h + (size_t)n * HID;
    float ss = 0.f, sd = 0.f;
#pragma unroll
    for (int c = 0; c < HID; ++c) {
        float v = r[c];
        ss += v * a_s[c];
        sd += v * a_d[c];
    }
    s_src[n] = ss;
    s_dst[n] = sd;
    float sv = leaky(ss + sd);   // self-loop edge score
    selfv[n] = sv;
    mkey[n] = fkey(sv);
}

// per-edge: score, segment max over dst
__global__ void edge_max(const int* __restrict__ src, const int* __restrict__ dst,
                         const float* __restrict__ s_src, const float* __restrict__ s_dst,
                         float* __restrict__ val_e, unsigned* __restrict__ mkey, int E) {
    int e = blockIdx.x * blockDim.x + threadIdx.x;
    if (e >= E) return;
    int d = dst[e];
    float v = leaky(s_src[src[e]] + s_dst[d]);
    val_e[e] = v;
    atomicMax(&mkey[d], fkey(v));
}

// per-node: decode max, ex_self = exp(self - m), den init
__global__ void node_den_init(const unsigned* __restrict__ mkey, float* __restrict__ selfv,
                              float* __restrict__ mf, float* __restrict__ den, int N) {
    int n = blockIdx.x * blockDim.x + threadIdx.x;
    if (n >= N) return;
    float m = kinv(mkey[n]);
    mf[n] = m;
    float ex = __expf(selfv[n] - m);
    selfv[n] = ex;   // now holds exp(self - m)
    den[n] = ex;     // softmax denominator seeded with self-loop
}

// per-edge: ex = exp(v - m[dst]); accumulate denominator
__global__ void edge_den(const int* __restrict__ dst, const float* __restrict__ mf,
                         float* __restrict__ val_e, float* __restrict__ den, int E) {
    int e = blockIdx.x * blockDim.x + threadIdx.x;
    if (e >= E) return;
    int d = dst[e];
    float ex = __expf(val_e[e] - mf[d]);
    val_e[e] = ex;   // now holds exp(v - m[dst])
    atomicAdd(&den[d], ex);
}

// out[n][c] = alpha_self * h[n][c]
__global__ void out_init(const float* __restrict__ h, const float* __restrict__ selfex,
                         const float* __restrict__ den, float* __restrict__ out, int total) {
    int i = blockIdx.x * blockDim.x + threadIdx.x;
    if (i >= total) return;
    int n = i >> 6;
    out[i] = (selfex[n] / den[n]) * h[i];
}

// out[dst][c] += (ex_e / den[dst]) * h[src][c]
__global__ void edge_agg(const int* __restrict__ src, const int* __restrict__ dst,
                         const float* __restrict__ h, const float* __restrict__ val_e,
                         const float* __restrict__ den, float* __restrict__ out, int total) {
    int i = blockIdx.x * blockDim.x + threadIdx.x;
    if (i >= total) return;
    int e = i >> 6, c = i & 63;
    int s = src[e], d = dst[e];
    float w = val_e[e] / den[d];
    atomicAdd(&out[(size_t)d * HID + c], w * h[(size_t)s * HID + c]);
}

// out = relu(out + b), in place
__global__ void finalize_bias_relu(float* __restrict__ out, const float* __restrict__ b, int total) {
    int i = blockIdx.x * blockDim.x + threadIdx.x;
    if (i >= total) return;
    float v = out[i] + b[i & 63];
    out[i] = v > 0.f ? v : 0.f;
}

// ---------------------------------------------------------------------------
// Edge MLP: score[e] = relu([h[src]|h[dst]|ea[e]] @ Wm1 + bm1) @ Wm2 + bm2
// block = 128 threads (4 waves) -> 64 edges. Wave w handles cols 16w..16w+15
// over 4 edge-tiles of 16 (B fragments reused 4x). K split into 3 static
// region loops so the hot loop has zero branches.
// ---------------------------------------------------------------------------
__global__ __launch_bounds__(128) void edge_mlp_wmma(
    const float* __restrict__ h, const int* __restrict__ src, const int* __restrict__ dst,
    const float* __restrict__ ea, const float* __restrict__ Wm1, const float* __restrict__ bm1,
    const float* __restrict__ Wm2, const float* __restrict__ bm2, float* __restrict__ score, int E) {
    __shared__ float sacc[64];
    const int lane  = threadIdx.x & 31;
    const int wave  = threadIdx.x >> 5;
    const int col0  = wave * 16;
    const int mrow  = lane & 15;
    const int khalf = (lane >> 4) << 1;
    const int e0    = blockIdx.x * 64;

    const float* hs[4];
    const float* hd[4];
    const float* er[4];
#pragma unroll
    for (int t = 0; t < 4; ++t) {
        int e = e0 + t * 16 + mrow;
        if (e >= E) e = E - 1;  // clamp (E is a multiple of 64 here anyway)
        hs[t] = h + (size_t)src[e] * HID;
        hd[t] = h + (size_t)dst[e] * HID;
        er[t] = ea + (size_t)e * 16;
    }
    const float* wcol = Wm1 + col0 + mrow;

    if (threadIdx.x < 64) sacc[threadIdx.x] = 0.f;
    __syncthreads();

    v8f acc[4] = {};
    // region 1: k in [0,64) -> h[src]
    for (int k0 = 0; k0 < 64; k0 += 4) {
        int k = k0 + khalf;
        v2f b;
        b.x = wcol[(size_t)k * HID];
        b.y = wcol[(size_t)(k + 1) * HID];
#pragma unroll
        for (int t = 0; t < 4; ++t) {
            v2f a = *(const v2f*)(hs[t] + k);
            acc[t] = WMMA_F32X4(a, b, acc[t]);
        }
    }
    // region 2: k in [64,128) -> h[dst]
    for (int k0 = 64; k0 < 128; k0 += 4) {
        int k = k0 + khalf;
        v2f b;
        b.x = wcol[(size_t)k * HID];
        b.y = wcol[(size_t)(k + 1) * HID];
#pragma unroll
        for (int t = 0; t < 4; ++t) {
            v2f a = *(const v2f*)(hd[t] + (k - 64));
            acc[t] = WMMA_F32X4(a, b, acc[t]);
        }
    }
    // region 3: k in [128,144) -> edge_attr
#pragma unroll
    for (int k0 = 128; k0 < 144; k0 += 4) {
        int k = k0 + khalf;
        v2f b;
        b.x = wcol[(size_t)k * HID];
        b.y = wcol[(size_t)(k + 1) * HID];
#pragma unroll
        for (int t = 0; t < 4; ++t) {
            v2f a = *(const v2f*)(er[t] + (k - 128));
            acc[t] = WMMA_F32X4(a, b, acc[t]);
        }
    }

    const float bias = bm1[col0 + mrow];
    const float w2   = Wm2[col0 + mrow];
    const int rbase  = (lane >> 4) * 8;
#pragma unroll
    for (int t = 0; t < 4; ++t) {
#pragma unroll
        for (int r = 0; r < 8; ++r) {
            float z = acc[t][r] + bias;
            z = z > 0.f ? z : 0.f;
            atomicAdd(&sacc[t * 16 + rbase + r], z * w2);  // ds_add_f32 reduce over 64 cols
        }
    }
    __syncthreads();
    if (threadIdx.x < 64) {
        int eo = e0 + threadIdx.x;
        if (eo < E) score[eo] = sacc[threadIdx.x] + bm2[0];
    }
}

// ---------------------------------------------------------------------------
extern "C" void kernel_launch(void* const* d_in, const int* in_sizes, int n_in,
                              void* d_out, int out_size, void* d_ws, size_t ws_size,
                              hipStream_t stream) {
    const float* x   = (const float*)d_in[0];       // [N,128]
    const int*   ei  = (const int*)d_in[1];         // [2,E]
    const float* ea  = (const float*)d_in[2];       // [E,16]
    const float* W1  = (const float*)d_in[3];
    const float* as1 = (const float*)d_in[4];
    const float* ad1 = (const float*)d_in[5];
    const float* b1  = (const float*)d_in[6];
    const float* W2  = (const float*)d_in[7];
    const float* as2 = (const float*)d_in[8];
    const float* ad2 = (const float*)d_in[9];
    const float* b2  = (const float*)d_in[10];
    const float* Wm1 = (const float*)d_in[11];
    const float* bm1 = (const float*)d_in[12];
    const float* Wm2 = (const float*)d_in[13];
    const float* bm2 = (const float*)d_in[14];

    const int N = in_sizes[0] / 128;
    const int E = in_sizes[1] / 2;
    const int IN_NODE = 128;
    const int* src = ei;
    const int* dst = ei + E;

    // workspace layout (floats)
    float*    hA    = (float*)d_ws;              // [N*64] gemm output per layer
    float*    hB    = hA + (size_t)N * HID;      // [N*64] layer-1 aggregate / layer-2 input
    float*    s_src = hB + (size_t)N * HID;      // [N]
    float*    s_dst = s_src + N;                 // [N]
    float*    selfv = s_dst + N;                 // [N]
    unsigned* mkey  = (unsigned*)(selfv + N);    // [N]
    float*    mf    = (float*)(mkey + N);        // [N]
    float*    den   = mf + N;                    // [N]
    float*    val_e = den + N;                   // [E]

    float* score = (float*)d_out;                // [E]
    float* hout  = score + E;                    // [N*64] final node features

    const int TB = 256;
    const int gN   = (N + TB - 1) / TB;
    const int gE   = (E + TB - 1) / TB;
    const int tNH  = N * HID;
    const int tEH  = E * HID;
    const int gNH  = (tNH + TB - 1) / TB;
    const int gEH  = (tEH + TB - 1) / TB;
    const int gGemm = (N + 63) / 64;
    const int gMlp  = (E + 63) / 64;

    // ---- GAT layer 1: input x [N,128] ----
    gemm_n64_wmma<<<gGemm, 128, 0, stream>>>(x, W1, hA, N, IN_NODE);
    node_dots<<<gN, TB, 0, stream>>>(hA, as1, ad1, s_src, s_dst, selfv, mkey, N);
    edge_max<<<gE, TB, 0, stream>>>(src, dst, s_src, s_dst, val_e, mkey, E);
    node_den_init<<<gN, TB, 0, stream>>>(mkey, selfv, mf, den, N);
    edge_den<<<gE, TB, 0, stream>>>(dst, mf, val_e, den, E);
    out_init<<<gNH, TB, 0, stream>>>(hA, selfv, den, hB, tNH);
    edge_agg<<<gEH, TB, 0, stream>>>(src, dst, hA, val_e, den, hB, tEH);
    finalize_bias_relu<<<gNH, TB, 0, stream>>>(hB, b1, tNH);   // hB = layer-1 output

    // ---- GAT layer 2: input hB [N,64], aggregate straight into d_out ----
    gemm_n64_wmma<<<gGemm, 128, 0, stream>>>(hB, W2, hA, N, HID);
    node_dots<<<gN, TB, 0, stream>>>(hA, as2, ad2, s_src, s_dst, selfv, mkey, N);
    edge_max<<<gE, TB, 0, stream>>>(src, dst, s_src, s_dst, val_e, mkey, E);
    node_den_init<<<gN, TB, 0, stream>>>(mkey, selfv, mf, den, N);
    edge_den<<<gE, TB, 0, stream>>>(dst, mf, val_e, den, E);
    out_init<<<gNH, TB, 0, stream>>>(hA, selfv, den, hout, tNH);
    edge_agg<<<gEH, TB, 0, stream>>>(src, dst, hA, val_e, den, hout, tEH);
    finalize_bias_relu<<<gNH, TB, 0, stream>>>(hout, b2, tNH); // hout = final h (output 2)

    // ---- edge-scoring MLP on original edges ----
    edge_mlp_wmma<<<gMlp, 128, 0, stream>>>(hout, src, dst, ea, Wm1, bm1, Wm2, bm2, score, E);
}